// _BaseEncoder_24747601559958
// MI455X (gfx1250) — compile-verified
//
#include <hip/hip_runtime.h>
#include <stdint.h>

// One tile = 256 input int32 elements -> 256*33 float32 outputs.
#define TILE  256
#define CH    33                   // 1 sign channel + 32 binary digits
#define QUADS (TILE * CH / 4)      // 2112 16-byte quads per tile (33792 B)

// gfx12+/CDNA5 cache-policy immediate: TH[2:0]; 1 == non-temporal.
#define CPOL_RT 0
#define CPOL_NT 1

// ---- CDNA5 async global<->LDS path (gfx1250), guarded for portability ----
#if defined(__HIP_DEVICE_COMPILE__)
#if __has_builtin(__builtin_amdgcn_global_load_async_to_lds_b32) && \
    __has_builtin(__builtin_amdgcn_global_store_async_from_lds_b128)
#define USE_ASYNC_LDS 1
#else
#define USE_ASYNC_LDS 0
#endif
#else
#define USE_ASYNC_LDS 0
#endif

#if USE_ASYNC_LDS
// Typed address-space pointers matching the builtin signatures
// (diagnosed from hipcc: b32 takes AS1/AS3 int*, b128 takes AS1/AS3 v4i32*).
typedef int v4i __attribute__((__vector_size__(16)));
typedef __attribute__((address_space(1))) int  gint;    // global int
typedef __attribute__((address_space(3))) int  sint;    // LDS int
typedef __attribute__((address_space(1))) v4i  gint4;   // global v4i32
typedef __attribute__((address_space(3))) v4i  sint4;   // LDS v4i32
#endif

__device__ __forceinline__ void wait_asynccnt0() {
#if defined(__HIP_DEVICE_COMPILE__) && defined(__gfx1250__)
#if __has_builtin(__builtin_amdgcn_s_wait_asynccnt)
  __builtin_amdgcn_s_wait_asynccnt(0);
#else
  asm volatile("s_wait_asynccnt 0x0" ::: "memory");
#endif
#endif
}

__global__ __launch_bounds__(TILE) void base2_encode_kernel(
    const int* __restrict__ x, float* __restrict__ out, int n) {
  // Output tile staged in LDS with EXACTLY the global memory layout, so the
  // writeback is one linear 33792-byte stream of b128 stores.
  __shared__ __align__(16) float s_out[TILE * CH];   // 33792 B
  __shared__ __align__(16) int   s_in[TILE];         // 1024 B

  const int t      = threadIdx.x;
  const int base   = blockIdx.x * TILE;
  const int remain = n - base;

  if (remain >= TILE) {
    // ---- Phase 0: stage inputs into LDS via the async data mover ----------
#if USE_ASYNC_LDS
    __builtin_amdgcn_global_load_async_to_lds_b32(
        (gint*)(x + base + t), (sint*)(s_in + t), 0, CPOL_RT);
    wait_asynccnt0();            // each wave waits for its own async loads
#else
    s_in[t] = x[base + t];
#endif
    __syncthreads();

    // ---- Phase 1: expand one element per lane into 33 floats in LDS -------
    // Row base stride = 33 dwords; gcd(33,64)==1 => conflict-free banks.
    const int      xv = s_in[t];
    const unsigned a  = (xv < 0) ? (0u - (unsigned)xv) : (unsigned)xv;
    float* row = &s_out[t * CH];
    row[0] = (xv < 0) ? 1.0f : 0.0f;               // sign channel
#pragma unroll
    for (int c = 1; c <= 32; ++c) {                // MSD-first base-2 digits
      row[c] = (float)((a >> (32 - c)) & 1u);      // NORM /(base-1) is no-op
    }
    __syncthreads();                               // all ds stores visible

    // ---- Phase 2: linear LDS -> global stream, 16 B per lane per op -------
    // Output (277 MB) is write-once and exceeds the 192 MB L2: stream it
    // with TH=NT so it doesn't evict reusable lines.
    float* gbase = out + (size_t)base * CH;        // tile base, 16B-aligned
#if USE_ASYNC_LDS
    for (int q = t; q < QUADS; q += TILE) {
      __builtin_amdgcn_global_store_async_from_lds_b128(
          (gint4*)(gbase + 4 * q), (sint4*)(s_out + 4 * q), 0, CPOL_NT);
    }
    wait_asynccnt0();   // S_ENDPGM also implies wait-idle; this is belt+braces
#else
    float4*       g4 = (float4*)gbase;
    const float4* s4 = (const float4*)s_out;
    for (int q = t; q < QUADS; q += TILE) g4[q] = s4[q];
#endif
  } else if (t < remain) {
    // Partial tail tile (dead for 4096*512 inputs, kept for generality):
    // direct scalar path, correctness over throughput.
    const int      xv = x[base + t];
    const unsigned a  = (xv < 0) ? (0u - (unsigned)xv) : (unsigned)xv;
    float* row = out + (size_t)(base + t) * CH;
    row[0] = (xv < 0) ? 1.0f : 0.0f;
#pragma unroll
    for (int c = 1; c <= 32; ++c) row[c] = (float)((a >> (32 - c)) & 1u);
  }
}

extern "C" void kernel_launch(void* const* d_in, const int* in_sizes, int n_in,
                              void* d_out, int out_size, void* d_ws, size_t ws_size,
                              hipStream_t stream) {
  (void)n_in; (void)out_size; (void)d_ws; (void)ws_size;
  const int* x   = (const int*)d_in[0];
  float*     out = (float*)d_out;
  const int  n   = in_sizes[0];                // 4096*512 = 2,097,152
  const int  blocks = (n + TILE - 1) / TILE;   // 8192 tiles
  base2_encode_kernel<<<blocks, TILE, 0, stream>>>(x, out, n);
}